// RangeRNN_85993835200589
// MI455X (gfx1250) — compile-verified
//
#include <hip/hip_runtime.h>
#include <math.h>

// Problem constants (match the JAX reference).
#define TT 512
#define BB 128
#define II 512
#define HH 512

// Block-tile geometry: 8 waves (2x4), each wave owns a 32x32 tile (4 WMMA accums).
#define MBLK 64
#define NBLK 128
#define KC   32   // K chunk staged in LDS per TDM transfer

typedef __attribute__((ext_vector_type(2))) float v2f;
typedef __attribute__((ext_vector_type(8))) float v8f;
typedef __attribute__((ext_vector_type(4))) unsigned int u32x4;
typedef __attribute__((ext_vector_type(4))) int i32x4;
typedef __attribute__((ext_vector_type(8))) int i32x8;

#if defined(__has_builtin)
#if __has_builtin(__builtin_amdgcn_tensor_load_to_lds) && \
    __has_builtin(__builtin_amdgcn_s_wait_tensorcnt)
#define USE_TDM 1
#else
#define USE_TDM 0
#endif
#else
#define USE_TDM 0
#endif

__device__ __forceinline__ unsigned lds_off(const void* p) {
    // LDS aperture addresses carry the LDS byte offset in the low 32 bits.
    return (unsigned)(size_t)p;
}

// Issue one TDM 2-D tile DMA: global (row-major, row_stride = tensor_cols elems,
// fp32) -> LDS, packed row-major [tile_rows][tile_cols].
// D# layout per CDNA5 ISA.
__device__ __forceinline__ void tdm_load_tile(unsigned lds_byte_off,
                                              const float* gptr,
                                              int tile_rows, int tile_cols,
                                              unsigned tensor_rows,
                                              unsigned tensor_cols) {
#if USE_TDM
    unsigned long long ga = (unsigned long long)(size_t)gptr;
    u32x4 g0;
    g0.x = 1u;                                     // count=1 (valid), user mode
    g0.y = lds_byte_off;                           // LDS dest (bytes)
    g0.z = (unsigned)ga;                           // global_addr[31:0]
    g0.w = (unsigned)((ga >> 32) & 0x01FFFFFFull)  // global_addr[56:32]
           | (2u << 30);                           // type = 2 ("image")
    i32x8 g1;
    g1[0] = 0x00020000;                                          // data_size = 4B
    g1[1] = (int)((tensor_cols & 0xFFFFu) << 16);                // tensor_dim0 lo16
    g1[2] = (int)((tensor_cols >> 16) | ((tensor_rows & 0xFFFFu) << 16));
    g1[3] = (int)((tensor_rows >> 16) | ((unsigned)tile_cols << 16)); // tile_dim0
    g1[4] = tile_rows & 0xFFFF;                                  // tile_dim1
    g1[5] = (int)tensor_cols;                                    // tensor_dim0_stride lo32
    g1[6] = 0;
    g1[7] = 0;
    i32x4 gz = {0, 0, 0, 0};
#if __clang_major__ >= 23
    i32x8 gz8 = {0, 0, 0, 0, 0, 0, 0, 0};
    __builtin_amdgcn_tensor_load_to_lds(g0, g1, gz, gz, gz8, 0);
#else
    __builtin_amdgcn_tensor_load_to_lds(g0, g1, gz, gz, 0);
#endif
#else
    (void)lds_byte_off; (void)gptr; (void)tile_rows; (void)tile_cols;
    (void)tensor_rows; (void)tensor_cols;
#endif
}

#if !USE_TDM
// Fallback: cooperative 256-thread global->LDS copy (emits ds_store_b128).
__device__ __forceinline__ void coop_load_tile(float* __restrict__ dst,
                                               const float* __restrict__ src,
                                               int rows, int src_stride) {
    const int total = rows * KC / 4;
    for (int idx = threadIdx.x; idx < total; idx += 256) {
        const int rr = (idx * 4) / KC;
        const int cc = (idx * 4) % KC;
        *(float4*)(dst + rr * KC + cc) =
            *(const float4*)(src + (size_t)rr * src_stride + cc);
    }
}
#endif

// One K-chunk of WMMA work for this wave's 32x32 tile.
// Per CDNA5 wave32 f32 WMMA layouts: lane r (0..15), hl = lane>>4:
//   A frag: A[r, k+2*hl+{0,1}]   B frag: W[n+r, k+2*hl+{0,1}] (W stored [N,K])
//   C/D:    VGPR i <-> row i+8*hl, col r
__device__ __forceinline__ void compute_chunk(const float* __restrict__ as,
                                              const float* __restrict__ bs,
                                              int wm, int wn, int hl, int r,
                                              v8f& c00, v8f& c01, v8f& c10, v8f& c11) {
    const float* a0p = as + (wm * 32 + r) * KC + 2 * hl;
    const float* a1p = a0p + 16 * KC;
    const float* b0p = bs + (wn * 32 + r) * KC + 2 * hl;
    const float* b1p = b0p + 16 * KC;
#pragma unroll
    for (int k = 0; k < KC; k += 4) {
        v2f a0 = *(const v2f*)(a0p + k);
        v2f a1 = *(const v2f*)(a1p + k);
        v2f b0 = *(const v2f*)(b0p + k);
        v2f b1 = *(const v2f*)(b1p + k);
        c00 = __builtin_amdgcn_wmma_f32_16x16x4_f32(false, a0, false, b0, (short)0, c00, false, false);
        c01 = __builtin_amdgcn_wmma_f32_16x16x4_f32(false, a0, false, b1, (short)0, c01, false, false);
        c10 = __builtin_amdgcn_wmma_f32_16x16x4_f32(false, a1, false, b0, (short)0, c10, false, false);
        c11 = __builtin_amdgcn_wmma_f32_16x16x4_f32(false, a1, false, b1, (short)0, c11, false, false);
    }
}

// Block GEMM: out[bm:bm+64, bn:bn+128] (+= variant per IS_STEP) of A * Wt^T,
// A row-major [a_rows, K], Wt row-major [w_rows, K] (i.e. already transposed).
//   IS_STEP = false: out = A*Wt^T + bias[col]              (xw projection)
//   IS_STEP = true : out = tanh(A*Wt^T + out)  (in place)  (recurrence step)
template <bool IS_STEP>
__device__ __forceinline__ void gemm_block(const float* __restrict__ A,
                                           const float* __restrict__ Wt,
                                           const float* __restrict__ bias,
                                           float* __restrict__ out,
                                           unsigned a_rows, unsigned w_rows,
                                           int bm, int bn, int K) {
    __shared__ float As[2][MBLK * KC];
    __shared__ float Bs[2][NBLK * KC];

    const int wave = threadIdx.x >> 5;
    const int wm = wave >> 2;          // 0..1
    const int wn = wave & 3;           // 0..3
    const int lane = threadIdx.x & 31;
    const int hl = lane >> 4;
    const int r = lane & 15;

    v8f c00 = {}, c01 = {}, c10 = {}, c11 = {};
    const int nch = K / KC;

#if USE_TDM
    // TDM double-buffered pipeline: wave 0 drives the DMAs, TENSORcnt tracks them.
    if (threadIdx.x < 32) {
        tdm_load_tile(lds_off(&As[0][0]), A + (size_t)bm * K, MBLK, KC, a_rows, (unsigned)K);
        tdm_load_tile(lds_off(&Bs[0][0]), Wt + (size_t)bn * K, NBLK, KC, w_rows, (unsigned)K);
    }
    for (int ch = 0; ch < nch; ++ch) {
        const int cur = ch & 1;
        if (threadIdx.x < 32) {
            if (ch + 1 < nch) {
                const int k = (ch + 1) * KC;
                tdm_load_tile(lds_off(&As[cur ^ 1][0]), A + (size_t)bm * K + k,
                              MBLK, KC, a_rows, (unsigned)K);
                tdm_load_tile(lds_off(&Bs[cur ^ 1][0]), Wt + (size_t)bn * K + k,
                              NBLK, KC, w_rows, (unsigned)K);
                __builtin_amdgcn_s_wait_tensorcnt(2);  // chunk `ch` resident
            } else {
                __builtin_amdgcn_s_wait_tensorcnt(0);  // last chunk resident
            }
        }
        __syncthreads();
        compute_chunk(&As[cur][0], &Bs[cur][0], wm, wn, hl, r, c00, c01, c10, c11);
        __syncthreads();   // buffer free before it is DMA-overwritten
    }
#else
    for (int ch = 0; ch < nch; ++ch) {
        const int k = ch * KC;
        coop_load_tile(&As[0][0], A + (size_t)bm * K + k, MBLK, K);
        coop_load_tile(&Bs[0][0], Wt + (size_t)bn * K + k, NBLK, K);
        __syncthreads();
        compute_chunk(&As[0][0], &Bs[0][0], wm, wn, hl, r, c00, c01, c10, c11);
        __syncthreads();
    }
    (void)a_rows; (void)w_rows;
#endif

    // Epilogue.
    const int col0 = bn + wn * 32 + r;
    const int col1 = col0 + 16;
    float bv0 = 0.0f, bv1 = 0.0f;
    if (!IS_STEP) { bv0 = bias[col0]; bv1 = bias[col1]; }
#pragma unroll
    for (int i = 0; i < 8; ++i) {
        const int row0 = bm + wm * 32 + i + 8 * hl;
        const int row1 = row0 + 16;
        float* p00 = out + (size_t)row0 * HH + col0;
        float* p01 = out + (size_t)row0 * HH + col1;
        float* p10 = out + (size_t)row1 * HH + col0;
        float* p11 = out + (size_t)row1 * HH + col1;
        if (IS_STEP) {
            *p00 = tanhf(c00[i] + *p00);
            *p01 = tanhf(c01[i] + *p01);
            *p10 = tanhf(c10[i] + *p10);
            *p11 = tanhf(c11[i] + *p11);
        } else {
            *p00 = c00[i] + bv0;
            *p01 = c01[i] + bv1;
            *p10 = c10[i] + bv0;
            *p11 = c11[i] + bv1;
        }
    }
}

// Phase 1: xw = x @ W_ih^T + b, written into the outputs region of d_out.
__global__ __launch_bounds__(256) void xw_gemm_kernel(
    const float* __restrict__ x, const float* __restrict__ w_ih,
    const float* __restrict__ bias, float* __restrict__ out) {
    const int nbn = HH / NBLK;                       // 4
    const int bm = (blockIdx.x / nbn) * MBLK;
    const int bn = (blockIdx.x % nbn) * NBLK;
    gemm_block<false>(x, w_ih, bias, out, TT * BB, HH, bm, bn, II);
}

// Phase 2: out_t = tanh(out_t + h_prev @ W_hh^T), in place. One launch per t.
__global__ __launch_bounds__(256) void rnn_step_kernel(
    const float* __restrict__ h_prev, const float* __restrict__ w_hh,
    float* __restrict__ out_t) {
    const int nbn = HH / NBLK;                       // 4
    const int bm = (blockIdx.x / nbn) * MBLK;
    const int bn = (blockIdx.x % nbn) * NBLK;
    gemm_block<true>(h_prev, w_hh, nullptr, out_t, BB, HH, bm, bn, HH);
}

// Zero fill for the h_{-1} = 0 initial state (lives in workspace).
__global__ void zero_kernel(float* __restrict__ p, int n) {
    int i = blockIdx.x * blockDim.x + threadIdx.x;
    if (i < n) p[i] = 0.0f;
}

extern "C" void kernel_launch(void* const* d_in, const int* in_sizes, int n_in,
                              void* d_out, int out_size, void* d_ws, size_t ws_size,
                              hipStream_t stream) {
    const float* x    = (const float*)d_in[0];   // [T,B,I]
    const float* w_ih = (const float*)d_in[1];   // [H,I]
    const float* w_hh = (const float*)d_in[2];   // [H,H]
    const float* bias = (const float*)d_in[3];   // [H]
    float* out = (float*)d_out;                  // [T*B*H] outputs + [B*H] h_final
    float* h0  = (float*)d_ws;                   // zero initial state, B*H floats

    (void)in_sizes; (void)n_in; (void)out_size; (void)ws_size;

    // h_{-1} = 0
    zero_kernel<<<(BB * HH + 255) / 256, 256, 0, stream>>>(h0, BB * HH);

    // Phase 1: (T*B/64) x (H/128) = 1024 x 4 = 4096 blocks.
    xw_gemm_kernel<<<(TT * BB / MBLK) * (HH / NBLK), 256, 0, stream>>>(
        x, w_ih, bias, out);

    // Phase 2: 512 stream-ordered steps; (B/64) x (H/128) = 8 blocks each.
    for (int t = 0; t < TT; ++t) {
        const float* h_prev = (t == 0) ? h0 : (out + (size_t)(t - 1) * BB * HH);
        float* out_t = out + (size_t)t * BB * HH;
        rnn_step_kernel<<<(BB / MBLK) * (HH / NBLK), 256, 0, stream>>>(
            h_prev, w_hh, out_t);
    }

    // h_final = outputs[T-1] -> tail of d_out.
    hipMemcpyAsync(out + (size_t)TT * BB * HH,
                   out + (size_t)(TT - 1) * BB * HH,
                   (size_t)BB * HH * sizeof(float),
                   hipMemcpyDeviceToDevice, stream);
}